// GroupNormalization_58806692216853
// MI455X (gfx1250) — compile-verified
//
#include <hip/hip_runtime.h>

typedef float v2f __attribute__((ext_vector_type(2)));
typedef float v8f __attribute__((ext_vector_type(8)));
typedef float f4  __attribute__((ext_vector_type(4)));

constexpr int NF = 64;   // features
constexpr int NG = 8;    // groups (group = channel % 8 due to reshape(-1, G))
constexpr int NB = 16;   // batches
constexpr float GEPS = 1e-8f;

constexpr int THREADS = 256;          // 8 waves of 32
constexpr int WAVES_PER_BLOCK = THREADS / 32;
constexpr int P1_BLOCKS = 768;

// ---------------------------------------------------------------------------
// Kernel 0: zero the global stats accumulators and find batch boundaries by
// binary search (batch_ids are sorted -> no counting pass over N needed).
// ---------------------------------------------------------------------------
__global__ void gn_init(const int* __restrict__ bids, int N, float* __restrict__ ws_f) {
    int t = threadIdx.x;
    if (t < 2 * NB * NG) ws_f[t] = 0.0f;            // 256 stat slots (sum, sumsq)
    int* bounds = (int*)(ws_f + 2 * NB * NG);       // 17 ints
    if (t <= NB) {
        int lo = 0, hi = N;                          // lower_bound(batch_ids, t)
        while (lo < hi) {
            int mid = (lo + hi) >> 1;
            if (bids[mid] < t) lo = mid + 1; else hi = mid;
        }
        bounds[t] = lo;
    }
}

// ---------------------------------------------------------------------------
// Pass 1: segmented sum / sumsq per (batch, group) via WMMA with a one-hot
// A-matrix.  Each wave processes quads of 4 rows x 64 channels.
//   A (16x4 f32): A[m,k] = (batch_id(row k) == m)      -> exact segmentation
//   B (4x16  f32): per-row, per-group-class data pre-reduced over the four
//                  16-channel blocks (all map to the same group: g = n % 8)
//   D (16x16 f32): [batch m] x [group-class n] partial sums, in registers
// Main loop is completely unguarded (full quads only); <=3 tail rows are
// handled once by wave 0 with the guarded path.
// ---------------------------------------------------------------------------
__global__ void gn_stats(const float* __restrict__ feats, const int* __restrict__ bids,
                         int N, float* __restrict__ gsum, float* __restrict__ gsq) {
    __shared__ float lsum[NB * NG];
    __shared__ float lsq [NB * NG];
    const int t = threadIdx.x;
    if (t < NB * NG) { lsum[t] = 0.0f; lsq[t] = 0.0f; }
    __syncthreads();

    const int lane = t & 31;
    const int half = lane >> 4;      // 0: lanes 0-15, 1: lanes 16-31
    const int n    = lane & 15;      // M for A-matrix, N (group-class) for B
    const int waveId     = blockIdx.x * WAVES_PER_BLOCK + (t >> 5);
    const int totalWaves = (int)gridDim.x * WAVES_PER_BLOCK;

    const v8f zero8 = {0.f,0.f,0.f,0.f,0.f,0.f,0.f,0.f};
    v8f accS = zero8;
    v8f accQ = zero8;

    const int fullQuads = N >> 2;

    for (int Q = waveId; Q < fullQuads; Q += totalWaves) {
        const int R = Q << 2;
        // A rows: VGPR j of lane-half h holds K = 2h + j
        const int bid0 = bids[R + 2 * half];
        const int bid1 = bids[R + 2 * half + 1];
        v2f a;
        a.x = (bid0 == n) ? 1.0f : 0.0f;
        a.y = (bid1 == n) ? 1.0f : 0.0f;

        // B rows: VGPR j of lane-half h holds K = h + 2j
        const float* p0 = feats + (size_t)(R + half)     * NF + n;
        const float* p1 = feats + (size_t)(R + 2 + half) * NF + n;
        const float x0 = p0[0], x1 = p0[16], x2 = p0[32], x3 = p0[48];
        const float y0 = p1[0], y1 = p1[16], y2 = p1[32], y3 = p1[48];

        v2f bs, bq;
        bs.x = (x0 + x1) + (x2 + x3);
        bs.y = (y0 + y1) + (y2 + y3);
        bq.x = fmaf(x0, x0, fmaf(x1, x1, fmaf(x2, x2, x3 * x3)));
        bq.y = fmaf(y0, y0, fmaf(y1, y1, fmaf(y2, y2, y3 * y3)));

        accS = __builtin_amdgcn_wmma_f32_16x16x4_f32(
            false, a, false, bs, (short)0, accS, false, false);
        accQ = __builtin_amdgcn_wmma_f32_16x16x4_f32(
            false, a, false, bq, (short)0, accQ, false, false);
    }

    // Tail: remaining N % 4 rows, processed once by wave 0 (guarded path).
    if ((N & 3) && waveId == 0) {
        const int R   = fullQuads << 2;
        const int rA0 = R + 2 * half, rA1 = rA0 + 1;
        const int bid0 = (rA0 < N) ? bids[rA0] : -1;
        const int bid1 = (rA1 < N) ? bids[rA1] : -1;
        v2f a;
        a.x = (bid0 == n) ? 1.0f : 0.0f;
        a.y = (bid1 == n) ? 1.0f : 0.0f;

        const int r0 = R + half, r1 = R + 2 + half;
        const float* p0 = feats + (size_t)r0 * NF + n;
        const float* p1 = feats + (size_t)r1 * NF + n;
        float x0 = 0.f, x1 = 0.f, x2 = 0.f, x3 = 0.f;
        float y0 = 0.f, y1 = 0.f, y2 = 0.f, y3 = 0.f;
        if (r0 < N) { x0 = p0[0]; x1 = p0[16]; x2 = p0[32]; x3 = p0[48]; }
        if (r1 < N) { y0 = p1[0]; y1 = p1[16]; y2 = p1[32]; y3 = p1[48]; }

        v2f bs, bq;
        bs.x = (x0 + x1) + (x2 + x3);
        bs.y = (y0 + y1) + (y2 + y3);
        bq.x = fmaf(x0, x0, fmaf(x1, x1, fmaf(x2, x2, x3 * x3)));
        bq.y = fmaf(y0, y0, fmaf(y1, y1, fmaf(y2, y2, y3 * y3)));

        accS = __builtin_amdgcn_wmma_f32_16x16x4_f32(
            false, a, false, bs, (short)0, accS, false, false);
        accQ = __builtin_amdgcn_wmma_f32_16x16x4_f32(
            false, a, false, bq, (short)0, accQ, false, false);
    }

    // D layout: lane l, VGPR v -> M = v + 8*(l>>4), N = l&15; group = N % 8.
    // Fold lanes n and n+8 (same batch, same group), then LDS-atomic merge.
#pragma unroll
    for (int v = 0; v < 8; ++v) {
        float s = accS[v];
        float q = accQ[v];
        s += __shfl_xor(s, 8, 32);
        q += __shfl_xor(q, 8, 32);
        if ((lane & 8) == 0) {
            const int m = v + 8 * half;     // batch
            const int g = n & 7;            // group
            atomicAdd(&lsum[m * NG + g], s);
            atomicAdd(&lsq [m * NG + g], q);
        }
    }
    __syncthreads();
    if (t < NB * NG) {
        atomicAdd(&gsum[t], lsum[t]);
        atomicAdd(&gsq [t], lsq [t]);
    }
}

// ---------------------------------------------------------------------------
// Finalize: mean / inverse-stddev table, laid out per batch as
// [mean g0..g7 | invstd g0..g7] so pass 2 can fetch aligned float4s.
// ---------------------------------------------------------------------------
__global__ void gn_finalize(const float* __restrict__ gsum, const float* __restrict__ gsq,
                            const int* __restrict__ bounds, float* __restrict__ table) {
    int t = threadIdx.x;
    if (t >= NB * NG) return;
    int b = t >> 3, g = t & 7;
    float cnt  = fmaxf(8.0f * (float)(bounds[b + 1] - bounds[b]), 1.0f);
    float mean = gsum[t] / cnt;
    float var  = fmaxf(gsq[t] / cnt - mean * mean, 0.0f);
    float inv  = rsqrtf(var + GEPS);
    table[b * 16 + g]     = mean;
    table[b * 16 + 8 + g] = inv;
}

// ---------------------------------------------------------------------------
// Pass 2: streaming normalize, iterated in REVERSE so it first touches the
// lines pass 1 most recently left in the 192 MB L2 (pass 1 sweeps forward).
// One float4 (4 channels) per thread, non-temporal B128 load/store.
// Channels 4q..4q+3 -> groups (4q+i)%8: q even -> g0..3, q odd -> g4..7.
// ---------------------------------------------------------------------------
__global__ void gn_apply(const float* __restrict__ feats, const int* __restrict__ bids,
                         const float* __restrict__ gamma, const float* __restrict__ beta,
                         const float* __restrict__ table, float* __restrict__ out, int N) {
    int lin   = blockIdx.x * blockDim.x + threadIdx.x;   // linear thread id
    int total = N * (NF / 4);
    if (lin >= total) return;
    int idx  = total - 1 - lin;                          // reversed float4 index
    int row  = idx >> 4;
    int q    = idx & 15;
    int qpar = q & 1;
    int b    = bids[row];

    f4 x    = __builtin_nontemporal_load((const f4*)feats + idx);
    f4 mean = *(const f4*)(table + b * 16 + qpar * 4);
    f4 inv  = *(const f4*)(table + b * 16 + 8 + qpar * 4);
    f4 gm   = *(const f4*)(gamma + q * 4);
    f4 bt   = *(const f4*)(beta  + q * 4);

    f4 y = (x - mean) * inv * gm + bt;
    __builtin_nontemporal_store(y, (f4*)out + idx);
}

// ---------------------------------------------------------------------------
extern "C" void kernel_launch(void* const* d_in, const int* in_sizes, int n_in,
                              void* d_out, int out_size, void* d_ws, size_t ws_size,
                              hipStream_t stream) {
    const float* feats = (const float*)d_in[0];
    const int*   bids  = (const int*)  d_in[1];
    const float* gamma = (const float*)d_in[2];
    const float* beta  = (const float*)d_in[3];
    float* out = (float*)d_out;
    const int N = in_sizes[0] / NF;

    float* ws_f   = (float*)d_ws;
    float* gsum   = ws_f;                 // 128 floats
    float* gsq    = ws_f + 128;           // 128 floats
    int*   bounds = (int*)(ws_f + 256);   // 17 ints
    float* table  = ws_f + 288;           // 256 floats (16 batches x 16)

    gn_init    <<<1, 256, 0, stream>>>(bids, N, ws_f);
    gn_stats   <<<P1_BLOCKS, THREADS, 0, stream>>>(feats, bids, N, gsum, gsq);
    gn_finalize<<<1, 128, 0, stream>>>(gsum, gsq, bounds, table);

    const int total4  = N * (NF / 4);
    const int blocks2 = (total4 + THREADS - 1) / THREADS;
    gn_apply   <<<blocks2, THREADS, 0, stream>>>(feats, bids, gamma, beta, table, out, N);
}